// SparseGather_63488206569806
// MI455X (gfx1250) — compile-verified
//
#include <hip/hip_runtime.h>
#include <stdint.h>

// Sparse block gather (sbnet): x[N=8,C=64,H=512,W=512] fp32 NCHW,
// indices[NB=4096,3] -> out[NB,16,16,64] NHWC blocks.
//
// Strategy (MI455X / gfx1250):
//  - One workgroup (256 thr = 8 waves) per block.
//  - Tensor Data Mover: single TENSOR_LOAD_TO_LDS per block pulls the
//    [c=64][h=16][w=16] tile into LDS (rows of 64B contiguous in memory),
//    with LDS padding of 1 dword per 256-dword c-plane so the transposed
//    LDS reads are bank-conflict free.
//  - All waves then write the 64KB output block fully coalesced
//    (contiguous NHWC) with b128 stores.

typedef uint32_t v4u __attribute__((ext_vector_type(4)));
typedef int      v8i __attribute__((ext_vector_type(8)));
typedef int      v4i __attribute__((ext_vector_type(4)));
typedef float    v4f __attribute__((ext_vector_type(4)));

#define C_    64
#define H_    512
#define W_    512
#define BH_   16
#define BW_   16
#define PLANE 257   // 256 dwords per c-plane + 1 pad dword (TDM pad_enable)

__global__ __launch_bounds__(256) void SparseGatherTDM_kernel(
    const float* __restrict__ x,
    const int*   __restrict__ idx,
    float*       __restrict__ out)
{
    __shared__ float tile[C_ * PLANE];   // 65792 bytes

    const int b = blockIdx.x;

    // Block coordinates; force scalar (descriptor lives in SGPRs).
    int n  = __builtin_amdgcn_readfirstlane(idx[3 * b + 0]);
    int by = __builtin_amdgcn_readfirstlane(idx[3 * b + 1]);
    int bx = __builtin_amdgcn_readfirstlane(idx[3 * b + 2]);

    if (threadIdx.x < 32u) {
        // Tile start address inside the tensor.
        const uint64_t gaddr = (uint64_t)(uintptr_t)x +
            4ull * ((uint64_t)n * (uint64_t)(C_ * H_ * W_) +
                    (uint64_t)(by * BH_) * (uint64_t)W_ +
                    (uint64_t)(bx * BW_));
        // Low 32 bits of a generic LDS pointer == workgroup LDS byte offset.
        const uint32_t lds_base = (uint32_t)(uintptr_t)(&tile[0]);

        // ---- D# group 0 (128b): count / lds_addr / global_addr / type ----
        v4u g0;
        g0[0] = 1u;                                  // count=1, user mode
        g0[1] = lds_base;                            // lds_addr [63:32]
        g0[2] = (uint32_t)gaddr;                     // global_addr[31:0]
        g0[3] = ((uint32_t)(gaddr >> 32) & 0x01FFFFFFu) | (2u << 30); // addr[56:32], type=2

        // ---- D# group 1 (256b) ----
        // w0: workgroup_mask=0, data_size=2 (4B), pad_enable=1,
        //     pad_interval=7 (256 dwords), pad_amount=0 (1 dword)
        v8i g1;
        g1[0] = (2 << 16) | (1 << 20) | (7 << 22);
        g1[1] = (int)((W_ & 0xFFFF) << 16);          // tensor_dim0[15:0] @ [63:48]
        g1[2] = (int)(((H_ & 0xFFFF) << 16) | (W_ >> 16)); // dim0 hi / dim1 lo
        g1[3] = (int)((BW_ << 16) | (H_ >> 16));     // dim1 hi / tile_dim0=16
        g1[4] = (int)((C_ << 16) | BH_);             // tile_dim1=16 / tile_dim2=64
        g1[5] = (int)W_;                             // tensor_dim0_stride lo32 = 512
        g1[6] = (int)(((H_ * W_) & 0xFFFF) << 16);   // stride0 hi16 / stride1 lo16
        g1[7] = (int)((H_ * W_) >> 16);              // tensor_dim1_stride[47:16] = 4

        // ---- D# group 2 (128b): tensor_dim2=64, no dim3/iterate ----
        v4i g2;
        g2[0] = C_;
        g2[1] = 0;
        g2[2] = 0;
        g2[3] = 0;

        // ---- D# group 3 (128b): unused dims ----
        v4i g3 = {0, 0, 0, 0};

        // Extra group required by this toolchain's 6-arg prototype
        // (beyond the architectural 4-group D#); zero-filled.
        v8i g4 = {0, 0, 0, 0, 0, 0, 0, 0};

        // amdgpu-toolchain / clang-23 form:
        // (v4u g0, v8i g1, v4i g2, v4i g3, v8i extra, i32 cpol)
        __builtin_amdgcn_tensor_load_to_lds(g0, g1, g2, g3, g4, 0);
        __builtin_amdgcn_s_wait_tensorcnt(0);
    }
    __syncthreads();

    // Write the 64KB output block: out[b, i, j, c] is fully contiguous.
    float* ob = out + (size_t)b * (size_t)(BH_ * BW_ * C_);
    const int t = (int)threadIdx.x;
    #pragma unroll
    for (int k = 0; k < 16; ++k) {
        const int o = t * 4 + k * 1024;   // float4 base index, o in [0,16384)
        const int c = o & 63;
        const int j = (o >> 6) & 15;
        const int i = o >> 10;
        const int l = i * 16 + j;         // position within a c-plane
        v4f v;
        v.x = tile[(c + 0) * PLANE + l];
        v.y = tile[(c + 1) * PLANE + l];
        v.z = tile[(c + 2) * PLANE + l];
        v.w = tile[(c + 3) * PLANE + l];
        *(v4f*)(ob + o) = v;
    }
}

extern "C" void kernel_launch(void* const* d_in, const int* in_sizes, int n_in,
                              void* d_out, int out_size, void* d_ws, size_t ws_size,
                              hipStream_t stream) {
    const float* x   = (const float*)d_in[0];
    const int*   idx = (const int*)d_in[1];
    float*       out = (float*)d_out;

    const int nb = in_sizes[1] / 3;   // 4096 blocks
    SparseGatherTDM_kernel<<<nb, 256, 0, stream>>>(x, idx, out);
}